// LSTM_hid_2688649527375
// MI455X (gfx1250) — compile-verified
//
#include <hip/hip_runtime.h>
#include <hip/hip_bf16.h>

// ---------------------------------------------------------------------------
// Problem constants (from reference): H=10, IN=2, T=1024, P=16, TAG=2
// Gates per direction = 4*H = 40.
// ---------------------------------------------------------------------------
constexpr int T_LEN = 1024;
constexpr int GATES = 40;   // 4*H
constexpr int HID   = 10;
constexpr int P_N   = 16;

// d_ws layout (float offsets). Total ~5.57 MB.
constexpr size_t NODE_F_OFF = 0;                                   // [T,40]
constexpr size_t NODE_B_OFF = NODE_F_OFF + (size_t)T_LEN * GATES;  // [T,40]
constexpr size_t EDGE_F_OFF = NODE_B_OFF + (size_t)T_LEN * GATES;  // [P,T,40]
constexpr size_t EDGE_B_OFF = EDGE_F_OFF + (size_t)P_N * T_LEN * GATES;
constexpr size_t RES_OFF    = EDGE_B_OFF + (size_t)P_N * T_LEN * GATES; // [40]

typedef float v2f __attribute__((ext_vector_type(2)));
typedef float v8f __attribute__((ext_vector_type(8)));

// ---------------------------------------------------------------------------
// Kernel 1: batched input projection  pre[t][j] = Wih[j,:] @ x_t + b[j]
// via V_WMMA_F32_16X16X4_F32 (M=16 rows of t, N=16 gate cols, K=4 exact).
//
// A (16x4 f32, 2 VGPR): lanes 0-15 hold M=0..15 with K0 (v0), K1 (v1);
//                       lanes 16-31 hold K2 (v0), K3 (v1).     [ISA 7.12.2]
// B (4x16 f32, 2 VGPR): rows striped across lanes; v0 = K0 (lanes 0-15) |
//                       K2 (lanes 16-31); v1 = K1 | K3 (mirrors C/D split).
// C/D (16x16 f32, 8 VGPR): vgpr i -> M = i (+8 for upper lanes), N = lane&15.
// Bias is folded into C (single WMMA covers all of K).
// Tile decode: arr0 node_f (192 tiles), arr1 node_b (192),
//              arr2 edge_f (16*192), arr3 edge_b (16*192). 6528 total.
// ---------------------------------------------------------------------------
__global__ __launch_bounds__(32)
void preact_wmma(const float* __restrict__ x,       // [T,2]
                 const float* __restrict__ others,  // [P,T,2]
                 const float* __restrict__ nWihF, const float* __restrict__ nbF,
                 const float* __restrict__ nWihB, const float* __restrict__ nbB,
                 const float* __restrict__ eWihF, const float* __restrict__ ebF,
                 const float* __restrict__ eWihB, const float* __restrict__ ebB,
                 float* __restrict__ ws) {
  const int tile = blockIdx.x;
  const int lane = threadIdx.x;
  const int nl   = lane & 15;
  const int hi   = lane >> 4;    // 0: K0/K1 half, 1: K2/K3 half

  int arr, local, p = 0;
  if (tile < 192)       { arr = 0; local = tile; }
  else if (tile < 384)  { arr = 1; local = tile - 192; }
  else if (tile < 3456) { arr = 2; local = tile - 384;  p = local / 192; local %= 192; }
  else                  { arr = 3; local = tile - 3456; p = local / 192; local %= 192; }
  const int m0 = (local / 3) * 16;   // timestep tile base
  const int n0 = (local % 3) * 16;   // gate tile base (0,16,32)

  const float* Wih; const float* bias; float* out; bool isEdge;
  if (arr == 0)      { Wih = nWihF; bias = nbF; out = ws + NODE_F_OFF; isEdge = false; }
  else if (arr == 1) { Wih = nWihB; bias = nbB; out = ws + NODE_B_OFF; isEdge = false; }
  else if (arr == 2) { Wih = eWihF; bias = ebF; out = ws + EDGE_F_OFF + (size_t)p * T_LEN * GATES; isEdge = true; }
  else               { Wih = eWihB; bias = ebB; out = ws + EDGE_B_OFF + (size_t)p * T_LEN * GATES; isEdge = true; }

  const int row = m0 + nl;           // timestep index for A
  v2f A;
  if (isEdge) {
    // xe_t = [x_t(2), other_p_t(2)]  -> K0,K1 from x ; K2,K3 from other
    const float* src = hi ? (others + ((size_t)p * T_LEN + row) * 2) : (x + (size_t)row * 2);
    A[0] = src[0];
    A[1] = src[1];
  } else {
    // node input is K=2, pad K2=K3=0
    A[0] = hi ? 0.0f : x[(size_t)row * 2 + 0];
    A[1] = hi ? 0.0f : x[(size_t)row * 2 + 1];
  }

  const int  n   = n0 + nl;          // gate index for B / C / D column
  const bool nok = (n < GATES);
  v2f B;
  if (isEdge) {                       // Wih: [40,4], B[k][n] = Wih[n][k]
    B[0] = nok ? Wih[n * 4 + (hi ? 2 : 0)] : 0.0f;
    B[1] = nok ? Wih[n * 4 + (hi ? 3 : 1)] : 0.0f;
  } else {                            // Wih: [40,2]
    B[0] = (nok && !hi) ? Wih[n * 2 + 0] : 0.0f;
    B[1] = (nok && !hi) ? Wih[n * 2 + 1] : 0.0f;
  }

  const float bv = nok ? bias[n] : 0.0f;
  v8f C;
#pragma unroll
  for (int i = 0; i < 8; ++i) C[i] = bv;

  // 8 args: (neg_a, A, neg_b, B, c_mod, C, reuse_a, reuse_b)
  v8f D = __builtin_amdgcn_wmma_f32_16x16x4_f32(
      false, A, false, B, (short)0, C, false, false);

  if (nok) {
    const int mb = m0 + hi * 8;
#pragma unroll
    for (int i = 0; i < 8; ++i)
      out[(size_t)(mb + i) * GATES + n] = D[i];
  }
}

// ---------------------------------------------------------------------------
// Kernel 2: the three sequential recurrence chains, one wave32 each.
// Lane j computes gate j (lanes 0-7 also compute gates 32+j).
// Full h replicated in every lane; c lives in lanes 0-9 (lane == hidden unit).
// Gate order i,f,g,o (PyTorch): unit u uses g[u], g[10+u], g[20+u], g[30+u].
// ---------------------------------------------------------------------------
__device__ __forceinline__ float sig_(float v) { return 1.0f / (1.0f + __expf(-v)); }

__device__ __forceinline__ void lstm_step(const float* __restrict__ preRow, int lane,
                                          const float* w, const float* w2,
                                          float* h, float& c) {
  float g1 = preRow[lane];                               // gates 0..31
  float g2 = (lane < 8) ? preRow[32 + lane] : 0.0f;      // gates 32..39
#pragma unroll
  for (int k = 0; k < HID; ++k) { g1 += w[k] * h[k]; g2 += w2[k] * h[k]; }

  // gather the 4 gates for hidden unit u = lane (meaningful for lanes 0-9)
  const float gi  = g1;                                   // g[u]        (own lane)
  const float gf  = __shfl(g1, (lane + 10) & 31, 32);     // g[10+u]
  const float gg  = __shfl(g1, (lane + 20) & 31, 32);     // g[20+u]
  const float go1 = __shfl(g1, (lane + 30) & 31, 32);     // g[30+u], u<2
  const float go2 = __shfl(g2, (lane >= 2) ? (lane - 2) : 0, 32); // g[32+(u-2)]
  const float go  = (lane < 2) ? go1 : go2;

  const float i_ = sig_(gi);
  const float f_ = sig_(gf);
  const float t_ = tanhf(gg);
  const float o_ = sig_(go);
  const float cn = f_ * c + i_ * t_;
  const float hn = o_ * tanhf(cn);
  c = cn;                                                 // valid in lanes 0-9
#pragma unroll
  for (int u = 0; u < HID; ++u) h[u] = __shfl(hn, u, 32); // broadcast new h
}

__device__ __forceinline__ void load_whh(const float* __restrict__ Whh, int lane,
                                         float* w, float* w2) {
#pragma unroll
  for (int k = 0; k < HID; ++k) {
    w[k]  = Whh[lane * HID + k];                               // rows 0..31
    w2[k] = (lane < 8) ? Whh[(32 + lane) * HID + k] : 0.0f;    // rows 32..39
  }
}

__global__ __launch_bounds__(32)
void lstm_chains(const float* __restrict__ nodeWhhF,
                 const float* __restrict__ edgeWhhF,
                 const float* __restrict__ edgeWhhB,
                 float* __restrict__ ws) {
  const int lane = threadIdx.x;
  const float* nodePreF = ws + NODE_F_OFF;
  const float* nodePreB = ws + NODE_B_OFF;
  const float* edgePreF = ws + EDGE_F_OFF;
  const float* edgePreB = ws + EDGE_B_OFF;
  float* res = ws + RES_OFF;

  float w[HID], w2[HID], h[HID];
  float c = 0.0f;

  if (blockIdx.x == 0) {
    // ---- chain A: node forward full scan, then node backward single step ----
    load_whh(nodeWhhF, lane, w, w2);
#pragma unroll
    for (int u = 0; u < HID; ++u) h[u] = 0.0f;
    for (int t = 0; t < T_LEN; ++t) {
      const int tp = (t + 8 < T_LEN) ? (t + 8) : (T_LEN - 1);
      __builtin_prefetch(nodePreF + (size_t)tp * GATES + lane, 0, 1);
      lstm_step(nodePreF + (size_t)t * GATES, lane, w, w2, h, c);
    }
    if (lane == 0) {
#pragma unroll
      for (int u = 0; u < HID; ++u) res[u] = h[u];
    }
    // backward: single step from zero state (h0=0 => Whh term vanishes)
#pragma unroll
    for (int u = 0; u < HID; ++u) h[u] = 0.0f;
    c = 0.0f;
    lstm_step(nodePreB + (size_t)(T_LEN - 1) * GATES, lane, w, w2, h, c);
    if (lane == 0) {
#pragma unroll
      for (int u = 0; u < HID; ++u) res[10 + u] = h[u];
    }
  } else if (blockIdx.x == 1) {
    // ---- chain B: edge forward, state carried across 16 persons ----
    load_whh(edgeWhhF, lane, w, w2);
#pragma unroll
    for (int u = 0; u < HID; ++u) h[u] = 0.0f;
    for (int p = 0; p < P_N; ++p) {
      const float* base = edgePreF + (size_t)p * T_LEN * GATES;
      for (int t = 0; t < T_LEN; ++t) {
        const int tp = (t + 8 < T_LEN) ? (t + 8) : (T_LEN - 1);
        __builtin_prefetch(base + (size_t)tp * GATES + lane, 0, 1);
        lstm_step(base + (size_t)t * GATES, lane, w, w2, h, c);
      }
    }
    if (lane == 0) {
#pragma unroll
      for (int u = 0; u < HID; ++u) res[20 + u] = h[u];
    }
  } else {
    // ---- chain C: edge backward, persons 0..14 full, person 15 one step ----
    load_whh(edgeWhhB, lane, w, w2);
#pragma unroll
    for (int u = 0; u < HID; ++u) h[u] = 0.0f;
    for (int p = 0; p < P_N - 1; ++p) {
      const float* base = edgePreB + (size_t)p * T_LEN * GATES;
      for (int t = T_LEN - 1; t >= 0; --t) {
        const int tp = (t >= 8) ? (t - 8) : 0;
        __builtin_prefetch(base + (size_t)tp * GATES + lane, 0, 1);
        lstm_step(base + (size_t)t * GATES, lane, w, w2, h, c);
      }
    }
    lstm_step(edgePreB + ((size_t)(P_N - 1) * T_LEN + (T_LEN - 1)) * GATES,
              lane, w, w2, h, c);
    if (lane == 0) {
#pragma unroll
      for (int u = 0; u < HID; ++u) res[30 + u] = h[u];
    }
  }
}

// ---------------------------------------------------------------------------
// Kernel 3: tag[r] = lin_b[r] + sum_j (node20[j] + edge20[j]) * lin_W[r][j]
// ---------------------------------------------------------------------------
__global__ __launch_bounds__(32)
void finalize(const float* __restrict__ res, const float* __restrict__ linW,
              const float* __restrict__ linb, float* __restrict__ out) {
  const int r = threadIdx.x;
  if (r < 2) {
    float acc = linb[r];
#pragma unroll
    for (int j = 0; j < 20; ++j)
      acc += (res[j] + res[20 + j]) * linW[r * 20 + j];
    out[r] = acc;
  }
}

// ---------------------------------------------------------------------------
extern "C" void kernel_launch(void* const* d_in, const int* in_sizes, int n_in,
                              void* d_out, int out_size, void* d_ws, size_t ws_size,
                              hipStream_t stream) {
  (void)in_sizes; (void)n_in; (void)out_size; (void)ws_size;
  const float* self_pose = (const float*)d_in[0];   // [1,T,2]
  const float* others    = (const float*)d_in[1];   // [1,P,T,2]
  const float* nWihF = (const float*)d_in[2];
  const float* nWhhF = (const float*)d_in[3];
  const float* nbF   = (const float*)d_in[4];
  const float* nWihB = (const float*)d_in[5];
  const float* nWhhB = (const float*)d_in[6];  (void)nWhhB; // h0=0 => unused
  const float* nbB   = (const float*)d_in[7];
  const float* eWihF = (const float*)d_in[8];
  const float* eWhhF = (const float*)d_in[9];
  const float* ebF   = (const float*)d_in[10];
  const float* eWihB = (const float*)d_in[11];
  const float* eWhhB = (const float*)d_in[12];
  const float* ebB   = (const float*)d_in[13];
  const float* linW  = (const float*)d_in[14];
  const float* linb  = (const float*)d_in[15];
  float* ws  = (float*)d_ws;
  float* out = (float*)d_out;

  // 1) WMMA input projections for all four preactivation streams
  preact_wmma<<<6528, 32, 0, stream>>>(self_pose, others,
                                       nWihF, nbF, nWihB, nbB,
                                       eWihF, ebF, eWihB, ebB, ws);
  // 2) three concurrent sequential chains (1 wave32 each)
  lstm_chains<<<3, 32, 0, stream>>>(nWhhF, eWhhF, eWhhB, ws);
  // 3) final linear -> 2 output floats
  finalize<<<1, 32, 0, stream>>>(ws + RES_OFF, linW, linb, out);
}